// Decoder_83107617177925
// MI455X (gfx1250) — compile-verified
//
#include <hip/hip_runtime.h>
#include <stdint.h>

#ifndef __has_builtin
#define __has_builtin(x) 0
#endif

// ---------------- problem constants ----------------
#define TPB       256                 // threads per block, 8 waves (wave32)
#define NUM_T     200                 // time points (incl. t=0)
#define T_CHUNK   8                   // time points staged per LDS flush
#define N_CHUNK   (NUM_T / T_CHUNK)   // 25
#define ROW       (T_CHUNK * 3)       // 24 floats per system per chunk (96 B, 16B-aligned)
#define ROW_F4    (ROW / 4)           // 6 float4s per system per chunk
#define ROW_PAD   28                  // 112 B row stride: 16B-aligned, 2-way bank conflict max
#define SUBSTEPS  8

// ---------------- CDNA5 async-store helpers ----------------
// b32 builtin signature (confirmed by clang diagnostic): (AS1 int*, AS3 int*, Ii, Ii)
// b128 expected per the same pattern: (AS1 int4*, AS3 int4*, Ii, Ii)
typedef __attribute__((ext_vector_type(4))) int int4v;
typedef __attribute__((address_space(1))) int    gint_t;
typedef __attribute__((address_space(3))) int    lint_t;
typedef __attribute__((address_space(1))) int4v  gint4_t;
typedef __attribute__((address_space(3))) int4v  lint4_t;

// Stores 16 contiguous bytes LDS -> global (per lane), tracked by ASYNCcnt.
__device__ __forceinline__ void async_store_f32x4(float* gp, float* lp) {
#if __has_builtin(__builtin_amdgcn_global_store_async_from_lds_b128)
    __builtin_amdgcn_global_store_async_from_lds_b128(
        (gint4_t*)(void*)gp, (lint4_t*)(void*)lp, /*offset=*/0, /*cpol=*/0);
#elif __has_builtin(__builtin_amdgcn_global_store_async_from_lds_b32)
    #pragma unroll
    for (int i = 0; i < 4; ++i)
        __builtin_amdgcn_global_store_async_from_lds_b32(
            (gint_t*)(void*)(gp + i), (lint_t*)(void*)(lp + i), 0, 0);
#else
    #pragma unroll
    for (int i = 0; i < 4; ++i) gp[i] = lp[i];   // synchronous fallback
#endif
}

template <int N>
__device__ __forceinline__ void wait_asynccnt_le() {
#if __has_builtin(__builtin_amdgcn_s_wait_asynccnt)
    __builtin_amdgcn_s_wait_asynccnt(N);
#else
    asm volatile("s_wait_asynccnt %0" :: "i"(N) : "memory");
#endif
}

// ---------------- SIR dynamics (2-state: R = 1 - S - I) ----------------
__device__ __forceinline__ void sir_deriv(float beta, float gamma,
                                          float S, float I,
                                          float& dS, float& dI) {
    float p = beta * S * I;          // 2 mul
    dS = -p;
    dI = fmaf(-gamma, I, p);         // 1 fma
}

__global__ void __launch_bounds__(TPB)
sir_rk4_kernel(const float* __restrict__ params, float* __restrict__ out) {
    const int tid = threadIdx.x;
    const int b0  = blockIdx.x * TPB;
    const int b   = b0 + tid;

    // double-buffered trajectory staging: 2 * 256 * 28 * 4 = 56 KB
    __shared__ __align__(16) float buf[2][TPB][ROW_PAD];

    // coalesced float4 param load: (beta, gamma, S0, I0)
    const float4 p = reinterpret_cast<const float4*>(params)[b];
    const float beta  = p.x;
    const float gamma = p.y;
    float S = p.z;
    float I = p.w;

    const float tstep = 100.0f / 199.0f;   // linspace(0,100,200) step in fp32

    int   half  = 0;
    int   tslot = 0;
    int   chunk = 0;
    float tprev = 0.0f;

    for (int t = 0; t < NUM_T; ++t) {
        if (t > 0) {
            // dt matches fp32 linspace differencing: ts[t]-ts[t-1]
            // (compiler keeps this recurrence on the scalar float pipe)
            float tcur = (float)t * tstep;
            float dt   = tcur - tprev;
            tprev = tcur;

            const float h  = dt * (1.0f / (float)SUBSTEPS);
            const float hh = 0.5f * h;
            const float h6 = h * (1.0f / 6.0f);

            #pragma unroll
            for (int s = 0; s < SUBSTEPS; ++s) {
                float k1S, k1I, k2S, k2I, k3S, k3I, k4S, k4I;
                sir_deriv(beta, gamma, S, I, k1S, k1I);
                float aS = fmaf(hh, k1S, S), aI = fmaf(hh, k1I, I);
                sir_deriv(beta, gamma, aS, aI, k2S, k2I);
                float bS = fmaf(hh, k2S, S), bI = fmaf(hh, k2I, I);
                sir_deriv(beta, gamma, bS, bI, k3S, k3I);
                float cS = fmaf(h,  k3S, S), cI = fmaf(h,  k3I, I);
                sir_deriv(beta, gamma, cS, cI, k4S, k4I);
                float sS = fmaf(2.0f, k2S + k3S, k1S) + k4S;
                float sI = fmaf(2.0f, k2I + k3I, k1I) + k4I;
                S = fmaf(h6, sS, S);
                I = fmaf(h6, sI, I);
            }
        }

        // stage (S, I, R) for this time point in my LDS row
        float* row = &buf[half][tid][tslot * 3];
        row[0] = S;
        row[1] = I;
        row[2] = 1.0f - S - I;
        ++tslot;

        if (tslot == T_CHUNK) {
            tslot = 0;
            __syncthreads();   // buf[half] fully written by all waves

            // Cooperative, fully-coalesced flush of 256 systems x 24 floats
            // via wave-wide 512 B async b128 stores.
            // Global target: out[(b0+s)*600 + chunk*24 + f4*4]   (600 = 200*3)
            const size_t gbase = (size_t)b0 * (NUM_T * 3) + (size_t)chunk * ROW;
            #pragma unroll
            for (int k = 0; k < ROW_F4; ++k) {
                int idx = tid + k * TPB;        // 0 .. 256*6-1, consecutive per wave
                int s   = idx / ROW_F4;         // system within block
                int f4  = idx - s * ROW_F4;     // float4 within 24-float row
                float* gp = out + gbase + (size_t)s * (NUM_T * 3) + f4 * 4;
                float* lp = &buf[half][s][f4 * 4];
                async_store_f32x4(gp, lp);
            }

            // In-order async retirement: <=6 outstanding means the PREVIOUS
            // chunk's flush (which read buf[half^1]) has fully drained, so
            // after the barrier every wave may overwrite buf[half^1].
            wait_asynccnt_le<ROW_F4>();
            __syncthreads();

            half ^= 1;
            ++chunk;
        }
    }
    // S_ENDPGM performs an implicit wait-idle, covering the final flush.
}

extern "C" void kernel_launch(void* const* d_in, const int* in_sizes, int n_in,
                              void* d_out, int out_size, void* d_ws, size_t ws_size,
                              hipStream_t stream) {
    const float* params = (const float*)d_in[0];
    float* out = (float*)d_out;
    const int B = in_sizes[0] / 4;               // 65536 systems
    dim3 grid(B / TPB), block(TPB);
    hipLaunchKernelGGL(sir_rk4_kernel, grid, block, 0, stream, params, out);
}